// QDense4StatesUndirected_26817775796831
// MI455X (gfx1250) — compile-verified
//
#include <hip/hip_runtime.h>
#include <hip/hip_bf16.h>
#include <math.h>

// ---------------------------------------------------------------------------
// 12-qubit StronglyEntanglingLayers state-vector simulator for MI455X.
//  * One workgroup (128 threads = 4 wave32) per batch element.
//  * State (4096 complex64) lives in LDS, ping-pong 2 x 32KB.
//  * Per layer: 4 fused 3-wire gates, each a dense 16x16 real matrix applied
//    via chained V_WMMA_F32_16X16X4_F32 (K=16 in 4 steps), then the CNOT ring
//    collapsed into a single GF(2)-linear index permutation (gather pass).
//  * Normalization in the reference cancels (circuit is linear) -> skipped.
// ---------------------------------------------------------------------------

typedef float v2f __attribute__((ext_vector_type(2)));
typedef float v8f __attribute__((ext_vector_type(8)));

#define QDEPTH   8
#define WIRES    12
#define NSTATE   4096        // 2^12 amplitudes
#define NGROUPS  4           // 12 wires fused 3-at-a-time
#define TPB      128         // 4 waves

// ============================================================================
// Kernel 1: build the 4*QDEPTH fused gate matrices (16x16 real each) into ws.
//   Entry (l, g, row, col):  row = 2i+a, col = 2j+b  (a,b = re/im selector)
//   E = U_{3g}[i2][j2] * U_{3g+1}[i1][j1] * U_{3g+2}[i0][j0]   (complex)
//   R[2i+0][2j+0] =  Re E ; R[2i+0][2j+1] = -Im E
//   R[2i+1][2j+0] =  Im E ; R[2i+1][2j+1] =  Re E
// ============================================================================
__global__ void qsel_build_gates(const float* __restrict__ weights,
                                 float* __restrict__ gmat) {
    int e = blockIdx.x * blockDim.x + threadIdx.x;
    if (e >= QDEPTH * NGROUPS * 16 * 16) return;

    int col = e & 15;
    int row = (e >> 4) & 15;
    int g   = (e >> 8) & 3;
    int l   = e >> 10;

    int i = row >> 1, a = row & 1;   // output amplitude index / re-im
    int j = col >> 1, b = col & 1;   // input  amplitude index / re-im

    float er = 1.0f, ei = 0.0f;      // accumulated complex product
    #pragma unroll
    for (int q = 0; q < 3; ++q) {
        int wire = 3 * g + q;
        const float* wp = weights + (l * WIRES + wire) * 3;
        float phi = tanhf(wp[0]);
        float th  = tanhf(wp[1]);
        float om  = tanhf(wp[2]);
        float ct = cosf(0.5f * th), st = sinf(0.5f * th);
        float hp = 0.5f * (phi + om), hm = 0.5f * (phi - om);
        int ib = (i >> (2 - q)) & 1;
        int jb = (j >> (2 - q)) & 1;
        // Rot(phi,theta,omega) = RZ(om) RY(th) RZ(phi):
        //   U00 = e^{-i hp} c   U01 = -e^{+i hm} s
        //   U10 = e^{-i hm} s   U11 =  e^{+i hp} c
        float ur, ui;
        if (ib == 0 && jb == 0)      { ur =  cosf(hp) * ct; ui = -sinf(hp) * ct; }
        else if (ib == 0)            { ur = -cosf(hm) * st; ui = -sinf(hm) * st; }
        else if (jb == 0)            { ur =  cosf(hm) * st; ui = -sinf(hm) * st; }
        else                         { ur =  cosf(hp) * ct; ui =  sinf(hp) * ct; }
        float nr = er * ur - ei * ui;
        float ni = er * ui + ei * ur;
        er = nr; ei = ni;
    }
    float val = (a == 0) ? ((b == 0) ? er : -ei)
                         : ((b == 0) ? ei :  er);
    gmat[e] = val;
}

// ============================================================================
// Kernel 2: the simulator. grid.x = batch, block = 128 threads.
// Dynamic LDS: 2 * 8192 floats (ping-pong complex state, interleaved re/im).
// ============================================================================
__global__ __launch_bounds__(TPB) void qsel_apply(
        const float* __restrict__ x,
        const float* __restrict__ gmat,
        const int*   __restrict__ repsp,
        float*       __restrict__ out) {
    extern __shared__ float smem[];
    float* buf0 = smem;                 // 8192 floats
    float* buf1 = smem + 2 * NSTATE;    // 8192 floats

    const int b    = blockIdx.x;
    const int tid  = threadIdx.x;
    const int lane = tid & 31;
    const int wave = tid >> 5;

    // ---- load input (real), imaginary part = 0; norm cancels -> skip it ----
    const float* xb = x + (size_t)b * NSTATE;
    for (int n = tid; n < NSTATE; n += TPB) {
        float v = xb[n];
        buf0[2 * n]     = v;
        buf0[2 * n + 1] = 0.0f;
    }
    __syncthreads();

    const int reps = repsp[0];
    float* cur = buf0;
    float* alt = buf1;

    const int mrow  = lane & 15;          // A row / B-D column within tile
    const int koff  = (lane >> 4) << 1;   // lanes 16-31 carry K+2,K+3
    const int mbase = (lane >> 4) << 3;   // D rows: 0..7 or 8..15

    for (int rep = 0; rep < reps; ++rep) {
        for (int l = 0; l < QDEPTH; ++l) {
            // ================= 4 fused 3-wire gates (WMMA) =================
            for (int g = 0; g < NGROUPS; ++g) {
                // --- load this group's 16x16 A matrix into WMMA A layout ---
                const float* Ag = gmat + (((l << 2) + g) << 8);  // 256 floats
                v2f A[4];
                #pragma unroll
                for (int kk = 0; kk < 4; ++kk) {
                    A[kk].x = Ag[mrow * 16 + 4 * kk + koff + 0];
                    A[kk].y = Ag[mrow * 16 + 4 * kk + koff + 1];
                }

                const int s       = 9 - 3 * g;       // group LSB bit position
                const int lowmask = (1 << s) - 1;

                // 512 columns of 8 amplitudes -> 32 tiles of N=16, 8 per wave
                for (int tile = wave; tile < 32; tile += 4) {
                    const int col = (tile << 4) + mrow;
                    const int chi = ((col >> s) << (s + 3)) | (col & lowmask);

                    v8f C = {};   // K=16 done in one sweep -> start from zero
                    #pragma unroll
                    for (int kk = 0; kk < 4; ++kk) {
                        // B: K rows 4kk+koff, 4kk+koff+1 for column `col`
                        int k0 = 4 * kk + koff;
                        int k1 = k0 + 1;
                        v2f Bv;
                        Bv.x = cur[2 * (chi | ((k0 >> 1) << s)) + (k0 & 1)];
                        Bv.y = cur[2 * (chi | ((k1 >> 1) << s)) + (k1 & 1)];
#if __has_builtin(__builtin_amdgcn_wmma_f32_16x16x4_f32)
                        C = __builtin_amdgcn_wmma_f32_16x16x4_f32(
                                false, A[kk], false, Bv, (short)0, C,
                                false, false);
#else
                        // scalar fallback (should not be taken on gfx1250)
                        #pragma unroll
                        for (int r = 0; r < 8; ++r) {
                            C[r] += A[kk].x * Bv.x + A[kk].y * Bv.y;
                        }
#endif
                    }
                    // --- scatter D back: lane<16 -> rows 0..7, else 8..15 ---
                    #pragma unroll
                    for (int r = 0; r < 8; ++r) {
                        int m = mbase + r;
                        cur[2 * (chi | ((m >> 1) << s)) + (m & 1)] = C[r];
                    }
                }
                __syncthreads();
            }

            // ========== CNOT ring as one GF(2) permutation (gather) ========
            const int rr = (l % (WIRES - 1)) + 1;
            for (int m = tid; m < NSTATE; m += TPB) {
                int src = m;
                #pragma unroll
                for (int w = WIRES - 1; w >= 0; --w) {
                    int t = w + rr; if (t >= WIRES) t -= WIRES;
                    int pc = 11 - w;
                    int pt = 11 - t;
                    src ^= ((src >> pc) & 1) << pt;   // CNOT is self-inverse
                }
                alt[2 * m]     = cur[2 * src];
                alt[2 * m + 1] = cur[2 * src + 1];
            }
            __syncthreads();
            float* tswap = cur; cur = alt; alt = tswap;
        }
    }

    // ---- write complex64 output (interleaved re/im), coalesced ----
    float* ob = out + (size_t)b * (2 * NSTATE);
    for (int n = tid; n < NSTATE; n += TPB) {
        ob[2 * n]     = cur[2 * n];
        ob[2 * n + 1] = cur[2 * n + 1];
    }
}

// ============================================================================
extern "C" void kernel_launch(void* const* d_in, const int* in_sizes, int n_in,
                              void* d_out, int out_size, void* d_ws, size_t ws_size,
                              hipStream_t stream) {
    const float* x       = (const float*)d_in[0];   // (B,1,64,64) f32
    const float* weights = (const float*)d_in[1];   // (8,12,3) f32
    const int*   reps    = (const int*)d_in[2];     // scalar int
    float*       out     = (float*)d_out;           // (B,1,64,64) complex64
    float*       gmat    = (float*)d_ws;            // 8*4*16*16 f32 = 32 KB

    const int batch = in_sizes[0] / NSTATE;

    // Build fused gate matrices (8192 entries, one thread each).
    qsel_build_gates<<<dim3(32), dim3(256), 0, stream>>>(weights, gmat);

    // Simulate: one workgroup per batch state, 64 KB dynamic LDS ping-pong.
    qsel_apply<<<dim3(batch), dim3(TPB), 2 * 2 * NSTATE * sizeof(float),
                 stream>>>(x, gmat, reps, out);
}